// CKSAAP_28862180229304
// MI455X (gfx1250) — compile-verified
//
#include <hip/hip_runtime.h>

// CKSAAP pair-histogram embedding for MI455X (gfx1250).
//
// Roofline: 235 MB compulsory traffic @ 23.3 TB/s ~= 10 us; only ~134 MFLOP
// => scatter/memory bound. Main loop: ds_add_f32 LDS histogram scatter
// (conflict-free: 32 consecutive banks per wave), sequence row staged via
// gfx1250 async global->LDS DMA (ASYNCcnt), emb streamed with coalesced
// 128B-per-wave loads + global_prefetch (L2-resident: 128 MB emb < 192 MB L2).
// Epilogue: out = hist * (1/n) computed exactly in f32 on the matrix unit as
// D = A x (scale * shifted-identity) via v_wmma_f32_16x16x4_f32.

#define NUM_AA    20
#define NUM_PAIRS (NUM_AA * NUM_AA)   // 400
#define LSEQ      2048
#define THREADS   256
#define DHALF     32                  // D=64 split across gridDim.z = 2
#define DSTRIDE   33                  // padded LDS row stride (odd => no bank conflicts)

__global__ __launch_bounds__(THREADS) void cksaap_hist_kernel(
    const int* __restrict__ seq,
    const float* __restrict__ emb,
    float* __restrict__ out,
    int D, int kp1)
{
    __shared__ float hist[NUM_PAIRS * DSTRIDE];  // 52,800 B
    __shared__ int   sseq[LSEQ];                 //  8,192 B

    const int b     = blockIdx.x;
    const int t     = blockIdx.y;
    const int dh    = blockIdx.z;               // 0 or 1: which half of D
    const int tid   = threadIdx.x;
    const int gap   = t + 1;
    const int n     = LSEQ - gap;               // valid pair count
    const int dbase = dh * DHALF;

    const int*   seqb = seq + (size_t)b * LSEQ;
    const float* embb = emb + ((size_t)b * LSEQ) * (size_t)D + dbase;

    // ---------- Stage the sequence row into LDS (async DMA, ASYNCcnt) ----------
#if defined(__gfx1250__) && __has_builtin(__builtin_amdgcn_global_load_async_to_lds_b128)
    {
        typedef int v4i __attribute__((ext_vector_type(4)));
        typedef __attribute__((address_space(1))) v4i gv4i_t;   // global int4
        typedef __attribute__((address_space(3))) v4i lv4i_t;   // LDS int4
        for (int ch = tid; ch < (LSEQ * 4) / 16; ch += THREADS) {
            gv4i_t* gsrc = (gv4i_t*)(unsigned long long)(const void*)(seqb + ch * 4);
            lv4i_t* ldst = (lv4i_t*)(unsigned int)(unsigned long long)(const void*)(&sseq[ch * 4]);
            __builtin_amdgcn_global_load_async_to_lds_b128(gsrc, ldst, 0, 0);
        }
#if __has_builtin(__builtin_amdgcn_s_wait_asynccnt)
        __builtin_amdgcn_s_wait_asynccnt(0);
#else
        asm volatile("s_wait_asynccnt 0x0" ::: "memory");
#endif
    }
#else
    for (int j = tid; j < LSEQ; j += THREADS) sseq[j] = seqb[j];
#endif

    // ---------- Zero the LDS histogram ----------
    for (int j = tid; j < NUM_PAIRS * DSTRIDE; j += THREADS) hist[j] = 0.0f;
    __syncthreads();

    // ---------- Scatter-accumulate ----------
    // lane_d: one of 32 D-columns (consecutive LDS banks, coalesced 128B
    // global loads per wave); pg: 8 positions in flight across 8 waves.
    const int lane_d = tid & (DHALF - 1);
    const int pg     = tid >> 5;

    for (int i = pg; i < n; i += 8) {
        const int idx = sseq[i] * NUM_AA + sseq[i + gap];
        const float a = embb[(size_t)i * D + lane_d];
        const float c = embb[(size_t)(i + gap) * D + lane_d];
        if (i + 64 < n) {
            __builtin_prefetch(&embb[(size_t)(i + 64) * D + lane_d], 0, 1);
        }
        atomicAdd(&hist[idx * DSTRIDE + lane_d], 0.5f * (a + c));  // ds_add_f32
    }
    __syncthreads();

    // ---------- Normalize and write out ----------
    const float scale = 1.0f / (float)n;
    float* outb = out + (((size_t)b * kp1 + t) * NUM_PAIRS) * (size_t)D + dbase;

#if defined(__gfx1250__) && __has_builtin(__builtin_amdgcn_wmma_f32_16x16x4_f32)
    {
        // out_tile[16p x 16d] = hist_tile x (scale * I16), built from four
        // K=4 WMMAs with B = scale * shifted-identity. Exact in f32.
        typedef float v2f __attribute__((ext_vector_type(2)));
        typedef float v8f __attribute__((ext_vector_type(8)));
        const int lane = tid & 31;
        const int wv   = tid >> 5;          // 8 waves
        const int m    = lane & 15;         // A row / D column
        const int h    = lane >> 4;         // half select: K = j + 2h (A,B); M = r + 8h (D)
        // 25 pair-row tiles x 2 d-column tiles = 50 tiles of 16x16
        for (int tile = wv; tile < (NUM_PAIRS / 16) * (DHALF / 16); tile += THREADS / 32) {
            const int rt    = tile >> 1;
            const int ct    = tile & 1;
            const int pbase = rt * 16;
            v8f acc = {};
#pragma unroll
            for (int c0 = 0; c0 < 16; c0 += 4) {
                v2f A, B;
                A.x = hist[(pbase + m) * DSTRIDE + ct * 16 + c0 + 2 * h];
                A.y = hist[(pbase + m) * DSTRIDE + ct * 16 + c0 + 2 * h + 1];
                B.x = (m == (c0 + 2 * h))     ? scale : 0.0f;
                B.y = (m == (c0 + 2 * h + 1)) ? scale : 0.0f;
                acc = __builtin_amdgcn_wmma_f32_16x16x4_f32(
                          false, A, false, B, (short)0, acc, false, false);
            }
#pragma unroll
            for (int r = 0; r < 8; ++r) {
                outb[(size_t)(pbase + r + 8 * h) * D + ct * 16 + m] = acc[r];
            }
        }
    }
#else
    for (int p = pg; p < NUM_PAIRS; p += 8) {
        outb[(size_t)p * D + lane_d] = hist[p * DSTRIDE + lane_d] * scale;
    }
#endif
}

extern "C" void kernel_launch(void* const* d_in, const int* in_sizes, int n_in,
                              void* d_out, int out_size, void* d_ws, size_t ws_size,
                              hipStream_t stream) {
    (void)n_in; (void)d_ws; (void)ws_size;
    const int*   seq = (const int*)d_in[0];
    const float* emb = (const float*)d_in[1];
    // d_in[2] (k) lives on device; derive k+1 from output size instead so the
    // launch shape is host-computable and graph-capture safe.
    const int B   = in_sizes[0] / LSEQ;                       // 256
    const int D   = in_sizes[1] / in_sizes[0];                // 64
    const int kp1 = (int)((long long)out_size /
                          ((long long)B * NUM_PAIRS * (long long)D));  // 4
    dim3 grid((unsigned)B, (unsigned)kp1, (unsigned)(D / DHALF));
    cksaap_hist_kernel<<<grid, THREADS, 0, stream>>>(seq, emb, (float*)d_out, D, kp1);
}